// GNN_10969346474112
// MI455X (gfx1250) — compile-verified
//
#include <hip/hip_runtime.h>
#include <hip/hip_bf16.h>

#define N_NODES  100000
#define N_EDGES  1600000
#define N_GRAPHS 1000
#define HID      64
#define L_LAYERS 3

typedef __attribute__((ext_vector_type(2))) float v2f;
typedef __attribute__((ext_vector_type(8))) float v8f;

// ---------------------------------------------------------------------------
// Zero-fill workspace regions (graph-capture safe; no hipMemsetAsync needed).
// ---------------------------------------------------------------------------
__global__ void gnn_zero_kernel(float* __restrict__ p, int n) {
    int i = blockIdx.x * blockDim.x + threadIdx.x;
    int stride = gridDim.x * blockDim.x;
    for (; i < n; i += stride) p[i] = 0.0f;
}

// ---------------------------------------------------------------------------
// Edge scatter: agg[dst] += x[src].  16 threads per edge, float4 per thread.
// Feature rows are 256B contiguous -> fully coalesced gathers; scatter-adds
// land in the L2-resident 25.6MB agg buffer.
// ---------------------------------------------------------------------------
__global__ void gnn_edge_scatter(const float* __restrict__ x,
                                 const int*   __restrict__ ei,
                                 float*       __restrict__ agg) {
    long long tid = (long long)blockIdx.x * blockDim.x + threadIdx.x;
    int e = (int)(tid >> 4);
    if (e >= N_EDGES) return;
    int c = ((int)tid & 15) * 4;
    int s = ei[e];
    int d = ei[N_EDGES + e];
    const float4 v = *(const float4*)(x + (long long)s * HID + c);
    float* o = agg + (long long)d * HID + c;
    atomicAdd(o + 0, v.x);
    atomicAdd(o + 1, v.y);
    atomicAdd(o + 2, v.z);
    atomicAdd(o + 3, v.w);
}

// ---------------------------------------------------------------------------
// Node update via f32 WMMA:  xnext = relu(agg @ Wrel + brel + x @ Wroot)
// One wave per 16x16 output tile; K=64 swept as 16 steps of V_WMMA_F32_16X16X4_F32.
// Block = 256 threads = 8 waves = 2 M-tiles x 4 N-tiles. 6250 M-tiles exactly.
// ---------------------------------------------------------------------------
__global__ __launch_bounds__(256)
void gnn_node_update(const float* __restrict__ agg,
                     const float* __restrict__ x,
                     const float* __restrict__ Wrel,
                     const float* __restrict__ brel,
                     const float* __restrict__ Wroot,
                     float*       __restrict__ xnext) {
    const int lane = threadIdx.x & 31;
    const int wv   = threadIdx.x >> 5;
    const int mt   = blockIdx.x * 2 + (wv >> 2);   // M tile (16 nodes)
    const int nt   = wv & 3;                       // N tile (16 cols)
    const int row0 = mt * 16;
    const int col0 = nt * 16;
    const int l16  = lane & 15;
    const int khalf = (lane >> 4) * 2;             // lanes 16-31 hold K+2,K+3

    v8f acc = {};

    // agg @ Wrel
    #pragma unroll
    for (int kk = 0; kk < 16; ++kk) {
        const int kb = kk * 4 + khalf;
        v2f a, b;
        const float* ap = agg + (long long)(row0 + l16) * HID + kb;
        a[0] = ap[0];  a[1] = ap[1];
        const float* bp = Wrel + kb * HID + col0 + l16;
        b[0] = bp[0];  b[1] = bp[HID];
        acc = __builtin_amdgcn_wmma_f32_16x16x4_f32(false, a, false, b,
                                                    (short)0, acc, false, false);
    }
    // x @ Wroot
    #pragma unroll
    for (int kk = 0; kk < 16; ++kk) {
        const int kb = kk * 4 + khalf;
        v2f a, b;
        const float* ap = x + (long long)(row0 + l16) * HID + kb;
        a[0] = ap[0];  a[1] = ap[1];
        const float* bp = Wroot + kb * HID + col0 + l16;
        b[0] = bp[0];  b[1] = bp[HID];
        acc = __builtin_amdgcn_wmma_f32_16x16x4_f32(false, a, false, b,
                                                    (short)0, acc, false, false);
    }

    // Bias + ReLU + store.  C/D layout: VGPR r -> M=r (lanes 0-15), M=r+8 (16-31).
    const int mofs = (lane >> 4) * 8;
    const int col  = col0 + l16;
    const float bias = brel[col];
    #pragma unroll
    for (int r = 0; r < 8; ++r) {
        const int row = row0 + mofs + r;
        xnext[(long long)row * HID + col] = fmaxf(acc[r] + bias, 0.0f);
    }
}

// ---------------------------------------------------------------------------
// Per-graph pooling: pool[batch[node]] += x[node].  16 threads/node, float4.
// Pool buffer is 256KB -> L2-resident atomics.
// ---------------------------------------------------------------------------
__global__ void gnn_pool_add(const float* __restrict__ x,
                             const int*   __restrict__ batch,
                             float*       __restrict__ pool) {
    long long tid = (long long)blockIdx.x * blockDim.x + threadIdx.x;
    int node = (int)(tid >> 4);
    if (node >= N_NODES) return;
    int c = ((int)tid & 15) * 4;
    int g = batch[node];
    const float4 v = *(const float4*)(x + (long long)node * HID + c);
    float* o = pool + (long long)g * HID + c;
    atomicAdd(o + 0, v.x);
    atomicAdd(o + 1, v.y);
    atomicAdd(o + 2, v.z);
    atomicAdd(o + 3, v.w);
}

// ---------------------------------------------------------------------------
// Readout MLP: 1000 graphs, one block (64 threads) each. Trivial FLOPs.
// ---------------------------------------------------------------------------
__global__ __launch_bounds__(64)
void gnn_mlp(const float* __restrict__ pool,
             const float* __restrict__ W1, const float* __restrict__ b1,
             const float* __restrict__ W2, const float* __restrict__ b2,
             const float* __restrict__ W3, const float* __restrict__ b3,
             float* __restrict__ out) {
    __shared__ float sh[HID];
    __shared__ float h1[64];
    __shared__ float h2[32];
    const int g = blockIdx.x;
    const int t = threadIdx.x;

    sh[t] = pool[g * HID + t];
    __syncthreads();

    float acc = b1[t];
    #pragma unroll 8
    for (int k = 0; k < HID; ++k) acc += sh[k] * W1[k * 64 + t];
    h1[t] = fmaxf(acc, 0.0f);
    __syncthreads();

    if (t < 32) {
        float a2 = b2[t];
        #pragma unroll 8
        for (int k = 0; k < 64; ++k) a2 += h1[k] * W2[k * 32 + t];
        h2[t] = fmaxf(a2, 0.0f);
    }
    __syncthreads();

    if (t == 0) {
        float a3 = b3[0];
        #pragma unroll
        for (int k = 0; k < 32; ++k) a3 += h2[k] * W3[k];
        out[g] = a3;
    }
}

// ---------------------------------------------------------------------------
extern "C" void kernel_launch(void* const* d_in, const int* in_sizes, int n_in,
                              void* d_out, int out_size, void* d_ws, size_t ws_size,
                              hipStream_t stream) {
    const float* x0    = (const float*)d_in[0];
    const int*   ei    = (const int*)  d_in[1];
    const int*   batch = (const int*)  d_in[2];
    const float* Wrel  = (const float*)d_in[3];
    const float* brel  = (const float*)d_in[4];
    const float* Wroot = (const float*)d_in[5];
    const float* W1    = (const float*)d_in[6];
    const float* b1    = (const float*)d_in[7];
    const float* W2    = (const float*)d_in[8];
    const float* b2    = (const float*)d_in[9];
    const float* W3    = (const float*)d_in[10];
    const float* b3    = (const float*)d_in[11];
    float* out = (float*)d_out;

    const long long NF = (long long)N_NODES * HID;   // 6.4M floats
    float* agg  = (float*)d_ws;
    float* xA   = agg + NF;
    float* xB   = xA + NF;
    float* pool = xB + NF;                           // N_GRAPHS*HID floats

    // zero pool accumulator once
    gnn_zero_kernel<<<256, 256, 0, stream>>>(pool, N_GRAPHS * HID);

    const float* xcur = x0;
    float* bufs[2] = {xA, xB};

    const int scatter_blocks = (int)(((long long)N_EDGES * 16 + 255) / 256);
    const int pool_blocks    = (int)(((long long)N_NODES * 16 + 255) / 256);
    const int tile_blocks    = (N_NODES / 16) / 2;   // 3125 blocks, 8 waves each

    for (int l = 0; l < L_LAYERS; ++l) {
        gnn_zero_kernel<<<2048, 256, 0, stream>>>(agg, (int)NF);
        gnn_edge_scatter<<<scatter_blocks, 256, 0, stream>>>(xcur, ei, agg);
        float* xn = bufs[l & 1];
        gnn_node_update<<<tile_blocks, 256, 0, stream>>>(
            agg, xcur,
            Wrel + (long long)l * HID * HID,
            brel + (long long)l * HID,
            Wroot + (long long)l * HID * HID,
            xn);
        gnn_pool_add<<<pool_blocks, 256, 0, stream>>>(xn, batch, pool);
        xcur = xn;
    }

    gnn_mlp<<<N_GRAPHS, 64, 0, stream>>>(pool, W1, b1, W2, b2, W3, b3, out);
    (void)in_sizes; (void)n_in; (void)out_size; (void)ws_size;
}